// HeteroGCN_67585605370475
// MI455X (gfx1250) — compile-verified
//
#include <hip/hip_runtime.h>
#include <hip/hip_bf16.h>

#define CC 256            // CIN == CH == 256
#define NB 64             // batches
#define GN_EPS 1e-5f
#define NPB 16            // nodes per block in segment-sum

typedef float v2f __attribute__((ext_vector_type(2)));
typedef float v8f __attribute__((ext_vector_type(8)));

// ---------------- generic fill ----------------
__global__ void k_fill(float* __restrict__ p, float v, int n) {
  int i = blockIdx.x * blockDim.x + threadIdx.x;
  if (i < n) p[i] = v;
}

// ---------------- WMMA f32 GEMM: C[M,N] = A[M,K] @ B[K,N] (+bias) ----------------
// Row-major everywhere. One wave32 per 16x16 tile. Requires M%16==0, N%16==0, K%4==0.
__global__ void k_wmma_gemm(const float* __restrict__ A, const float* __restrict__ B,
                            const float* __restrict__ bias, float* __restrict__ C,
                            int M, int N, int K) {
  const int wavesPerBlock = blockDim.x >> 5;
  const int wave = blockIdx.x * wavesPerBlock + (threadIdx.x >> 5);
  const int tilesN = N >> 4;
  const int tM = (wave / tilesN) << 4;
  const int tN = (wave % tilesN) << 4;
  if (tM >= M) return;                    // wave-uniform exit; live waves have EXEC all-1
  const int lane = threadIdx.x & 31;
  const int half = lane >> 4;             // 0: lanes 0-15, 1: lanes 16-31
  const int l16  = lane & 15;
  const int k0   = half << 1;             // K sub-offset {0,2}

  v8f acc = {};
#if __has_builtin(__builtin_amdgcn_wmma_f32_16x16x4_f32)
  // A 16x4 frag: lane holds A[l16][k0], A[l16][k0+1]
  // B 4x16 frag: lane holds B[k0][l16], B[k0+1][l16]
  const float* arow = A + (size_t)(tM + l16) * K + k0;
  const float* bcol = B + (size_t)k0 * N + tN + l16;
  for (int kk = 0; kk < K; kk += 4) {
    v2f a, b;
    a.x = arow[kk];
    a.y = arow[kk + 1];
    b.x = bcol[(size_t)kk * N];
    b.y = bcol[(size_t)kk * N + N];
    acc = __builtin_amdgcn_wmma_f32_16x16x4_f32(false, a, false, b,
                                                (short)0, acc, false, false);
  }
#else
  // scalar fallback (should not be taken on gfx1250)
  for (int g = 0; g < 8; ++g) {
    int m = tM + g + (half << 3);
    float s = 0.f;
    for (int kk = 0; kk < K; ++kk) s += A[(size_t)m * K + kk] * B[(size_t)kk * N + tN + l16];
    acc[g] = s;
  }
#endif
  const float bv = bias ? bias[tN + l16] : 0.0f;
  // C/D layout: VGPR g -> row g (lanes 0-15) / row g+8 (lanes 16-31), col = l16
#pragma unroll
  for (int g = 0; g < 8; ++g) {
    int m = tM + g + (half << 3);
    C[(size_t)m * N + tN + l16] = acc[g] + bv;
  }
}

// ---------------- GCN degree / scatter ----------------
__global__ void k_deg_count(const int* __restrict__ dst, float* __restrict__ deg, int E) {
  int e = blockIdx.x * blockDim.x + threadIdx.x;
  if (e < E) atomicAdd(&deg[dst[e]], 1.0f);
}

__global__ void k_rsqrt_inplace(float* __restrict__ p, int n) {
  int i = blockIdx.x * blockDim.x + threadIdx.x;
  if (i < n) p[i] = rsqrtf(p[i]);
}

// one block (256 lanes) per edge; lane = channel; coalesced gather + atomic scatter
__global__ void k_gcn_scatter(const float* __restrict__ xw, const int* __restrict__ src,
                              const int* __restrict__ dst, const float* __restrict__ dinv,
                              float* __restrict__ agg) {
  int e = blockIdx.x;
  int c = threadIdx.x;
  int s = src[e], d = dst[e];
  float norm = dinv[s] * dinv[d];
  atomicAdd(&agg[(size_t)d * CC + c], norm * xw[(size_t)s * CC + c]);
}

__global__ void k_gcn_self_bias(const float* __restrict__ xw, const float* __restrict__ dinv,
                                const float* __restrict__ bias, float* __restrict__ agg, int n) {
  int i = blockIdx.x, c = threadIdx.x;
  float di = dinv[i];
  agg[(size_t)i * CC + c] += di * di * xw[(size_t)i * CC + c] + bias[c];
}

// ---------------- batched (per-graph) segment reductions ----------------
__global__ void k_batch_cnt(const int* __restrict__ batch, float* __restrict__ cnt, int n) {
  int i = blockIdx.x * blockDim.x + threadIdx.x;
  if (i < n) atomicAdd(&cnt[batch[i]], 1.0f);
}

// batch[] is sorted -> accumulate NPB nodes in registers, atomic only at segment edges
__global__ void k_batch_sum(const float* __restrict__ x, const int* __restrict__ batch,
                            float* __restrict__ out, int n, int squared) {
  int base = blockIdx.x * NPB;
  int c = threadIdx.x;
  int end = base + NPB; if (end > n) end = n;
  int cur = batch[base];
  float s = 0.0f;
  for (int i = base; i < end; ++i) {
    int b = batch[i];
    if (b != cur) { atomicAdd(&out[cur * CC + c], s); s = 0.0f; cur = b; }
    float v = x[(size_t)i * CC + c];
    s += squared ? v * v : v;
  }
  atomicAdd(&out[cur * CC + c], s);
}

__global__ void k_gn_center(float* __restrict__ x, const int* __restrict__ batch,
                            const float* __restrict__ sum, const float* __restrict__ cnt,
                            const float* __restrict__ alpha, int n) {
  int i = blockIdx.x, c = threadIdx.x;
  int b = batch[i];
  float m = sum[b * CC + c] / fmaxf(cnt[b], 1.0f);
  x[(size_t)i * CC + c] -= alpha[c] * m;
}

// y = relu( w * xc * rsqrt(var+eps) + b [+ residual] )
__global__ void k_gn_final(const float* __restrict__ xc, const int* __restrict__ batch,
                           const float* __restrict__ sumsq, const float* __restrict__ cnt,
                           const float* __restrict__ w, const float* __restrict__ bb,
                           const float* __restrict__ residual, float* __restrict__ y, int n) {
  int i = blockIdx.x, c = threadIdx.x;
  int b = batch[i];
  float var = sumsq[b * CC + c] / fmaxf(cnt[b], 1.0f);
  float v = w[c] * xc[(size_t)i * CC + c] * rsqrtf(var + GN_EPS) + bb[c];
  if (residual) v += residual[(size_t)i * CC + c];
  y[(size_t)i * CC + c] = fmaxf(v, 0.0f);
}

// ---------------- pooling ----------------
__global__ void k_pool_mean(const float* __restrict__ sum, const float* __restrict__ cnt,
                            float* __restrict__ pooled) {
  int b = blockIdx.x, c = threadIdx.x;
  pooled[b * 2 * CC + c] = sum[b * CC + c] / fmaxf(cnt[b], 1.0f);
}

// x >= 0 (post-ReLU) -> uint bit order == float order; pooled max region pre-zeroed
__global__ void k_batch_max(const float* __restrict__ x, const int* __restrict__ batch,
                            float* __restrict__ pooled, int n) {
  int i = blockIdx.x, c = threadIdx.x;
  int b = batch[i];
  unsigned int v = __float_as_uint(x[(size_t)i * CC + c]);
  atomicMax((unsigned int*)&pooled[b * 2 * CC + CC + c], v);
}

// ---------------- head ----------------
__global__ void k_ln_relu(float* __restrict__ h, const float* __restrict__ g,
                          const float* __restrict__ b) {
  __shared__ float red[CC];
  int r = blockIdx.x, c = threadIdx.x;
  float v = h[r * CC + c];
  red[c] = v; __syncthreads();
  for (int s = CC / 2; s > 0; s >>= 1) { if (c < s) red[c] += red[c + s]; __syncthreads(); }
  float mu = red[0] / CC; __syncthreads();
  float d = v - mu;
  red[c] = d * d; __syncthreads();
  for (int s = CC / 2; s > 0; s >>= 1) { if (c < s) red[c] += red[c + s]; __syncthreads(); }
  float var = red[0] / CC;
  float val = g[c] * d * rsqrtf(var + GN_EPS) + b[c];
  h[r * CC + c] = fmaxf(val, 0.0f);
}

__global__ void k_head_out(const float* __restrict__ h, const float* __restrict__ W2,
                           const float* __restrict__ b2, float* __restrict__ out) {
  __shared__ float r0[CC], r1[CC];
  int r = blockIdx.x, c = threadIdx.x;
  float v = h[r * CC + c];
  r0[c] = v * W2[c * 2 + 0];
  r1[c] = v * W2[c * 2 + 1];
  __syncthreads();
  for (int s = CC / 2; s > 0; s >>= 1) {
    if (c < s) { r0[c] += r0[c + s]; r1[c] += r1[c + s]; }
    __syncthreads();
  }
  if (c == 0) { out[r * 2 + 0] = r0[0] + b2[0]; out[r * 2 + 1] = r1[0] + b2[1]; }
}

// ---------------- attention fuse (reads po/go already in d_out) ----------------
__global__ void k_attention(float* __restrict__ out, const float* __restrict__ attW,
                            const float* __restrict__ attB) {
  int r = threadIdx.x;
  if (r >= NB) return;
  float comb[4] = { out[r * 2], out[r * 2 + 1], out[128 + r * 2], out[128 + r * 2 + 1] };
  float l0 = attB[0], l1 = attB[1];
#pragma unroll
  for (int j = 0; j < 4; ++j) { l0 += comb[j] * attW[j * 2]; l1 += comb[j] * attW[j * 2 + 1]; }
  float mx = fmaxf(l0, l1);
  float e0 = expf(l0 - mx), e1 = expf(l1 - mx);
  float inv = 1.0f / (e0 + e1);
  float a0 = e0 * inv, a1 = e1 * inv;
  out[256 + r * 2]     = a0;
  out[256 + r * 2 + 1] = a1;
  out[384 + r * 2]     = a0 * comb[0] + a1 * comb[2];
  out[384 + r * 2 + 1] = a0 * comb[1] + a1 * comb[3];
}

// ---------------- host-side per-type driver ----------------
static void run_gemm(const float* A, const float* B, const float* bias, float* C,
                     int M, int N, int K, hipStream_t st) {
  int tiles = (M / 16) * (N / 16);
  int blocks = (tiles + 3) / 4;           // 4 waves (128 threads) per block
  k_wmma_gemm<<<blocks, 128, 0, st>>>(A, B, bias, C, M, N, K);
}

static void run_type(const float* x, const int* src, const int* dst, const int* batch,
                     int n, int E,
                     const float* W1, const float* b1, const float* W2w, const float* b2w,
                     const float* n1w, const float* n1b, const float* n1a,
                     const float* n2w, const float* n2b, const float* n2a,
                     const float* hW1, const float* hb1, const float* hlng, const float* hlnb,
                     const float* hW2, const float* hb2,
                     float* h1, float* xw, float* agg, float* dinv, float* sum, float* cnt,
                     float* pooled, float* hh, float* out2, hipStream_t st) {
  int gb = (n + 255) / 256;
  // degrees -> dinv (self-loop +1 folded into init)
  k_fill<<<gb, 256, 0, st>>>(dinv, 1.0f, n);
  k_deg_count<<<(E + 255) / 256, 256, 0, st>>>(dst, dinv, E);
  k_rsqrt_inplace<<<gb, 256, 0, st>>>(dinv, n);
  // batch counts (shared by both layers + pooling)
  k_fill<<<1, NB, 0, st>>>(cnt, 0.0f, NB);
  k_batch_cnt<<<gb, 256, 0, st>>>(batch, cnt, n);

  int segBlocks = (n + NPB - 1) / NPB;
  int fb = (n * CC + 255) / 256;

  // ---- layer 1: conv -> graphnorm -> relu -> h1 ----
  run_gemm(x, W1, nullptr, xw, n, CC, CC, st);
  k_fill<<<fb, 256, 0, st>>>(agg, 0.0f, n * CC);
  k_gcn_scatter<<<E, CC, 0, st>>>(xw, src, dst, dinv, agg);
  k_gcn_self_bias<<<n, CC, 0, st>>>(xw, dinv, b1, agg, n);
  k_fill<<<(NB * CC) / 256, 256, 0, st>>>(sum, 0.0f, NB * CC);
  k_batch_sum<<<segBlocks, CC, 0, st>>>(agg, batch, sum, n, 0);
  k_gn_center<<<n, CC, 0, st>>>(agg, batch, sum, cnt, n1a, n);
  k_fill<<<(NB * CC) / 256, 256, 0, st>>>(sum, 0.0f, NB * CC);
  k_batch_sum<<<segBlocks, CC, 0, st>>>(agg, batch, sum, n, 1);
  k_gn_final<<<n, CC, 0, st>>>(agg, batch, sum, cnt, n1w, n1b, nullptr, h1, n);

  // ---- layer 2: conv -> graphnorm -> +h1 -> relu -> agg (h2) ----
  run_gemm(h1, W2w, nullptr, xw, n, CC, CC, st);
  k_fill<<<fb, 256, 0, st>>>(agg, 0.0f, n * CC);
  k_gcn_scatter<<<E, CC, 0, st>>>(xw, src, dst, dinv, agg);
  k_gcn_self_bias<<<n, CC, 0, st>>>(xw, dinv, b2w, agg, n);
  k_fill<<<(NB * CC) / 256, 256, 0, st>>>(sum, 0.0f, NB * CC);
  k_batch_sum<<<segBlocks, CC, 0, st>>>(agg, batch, sum, n, 0);
  k_gn_center<<<n, CC, 0, st>>>(agg, batch, sum, cnt, n2a, n);
  k_fill<<<(NB * CC) / 256, 256, 0, st>>>(sum, 0.0f, NB * CC);
  k_batch_sum<<<segBlocks, CC, 0, st>>>(agg, batch, sum, n, 1);
  k_gn_final<<<n, CC, 0, st>>>(agg, batch, sum, cnt, n2w, n2b, h1, agg, n);

  // ---- pool: [mean | max] -> pooled[NB, 2*CC] ----
  k_fill<<<(NB * 2 * CC) / 256, 256, 0, st>>>(pooled, 0.0f, NB * 2 * CC);
  k_fill<<<(NB * CC) / 256, 256, 0, st>>>(sum, 0.0f, NB * CC);
  k_batch_sum<<<segBlocks, CC, 0, st>>>(agg, batch, sum, n, 0);
  k_pool_mean<<<NB, CC, 0, st>>>(sum, cnt, pooled);
  k_batch_max<<<n, CC, 0, st>>>(agg, batch, pooled, n);

  // ---- head ----
  run_gemm(pooled, hW1, hb1, hh, NB, CC, 2 * CC, st);
  k_ln_relu<<<NB, CC, 0, st>>>(hh, hlng, hlnb);
  k_head_out<<<NB, CC, 0, st>>>(hh, hW2, hb2, out2);
}

extern "C" void kernel_launch(void* const* d_in, const int* in_sizes, int n_in,
                              void* d_out, int out_size, void* d_ws, size_t ws_size,
                              hipStream_t stream) {
  const float* x_p     = (const float*)d_in[0];
  const float* x_g     = (const float*)d_in[1];
  const int*   ei_p    = (const int*)d_in[2];
  const int*   ei_g    = (const int*)d_in[3];
  const int*   batch_p = (const int*)d_in[4];
  const int*   batch_g = (const int*)d_in[5];
  const float* P[40];
  for (int i = 6; i < 40 && i < n_in; ++i) P[i] = (const float*)d_in[i];

  const int NPn = in_sizes[0] / CC;
  const int NGn = in_sizes[1] / CC;
  const int E   = in_sizes[2] / 2;
  const int Nmax = NPn > NGn ? NPn : NGn;

  // workspace (floats), shared across the two sequential graph types
  float* ws = (float*)d_ws;
  size_t off = 0;
  float* h1     = ws + off; off += (size_t)Nmax * CC;
  float* xw     = ws + off; off += (size_t)Nmax * CC;
  float* agg    = ws + off; off += (size_t)Nmax * CC;
  float* dinv   = ws + off; off += (size_t)Nmax;
  float* sum    = ws + off; off += (size_t)NB * CC;
  float* cnt    = ws + off; off += NB;
  float* pooled = ws + off; off += (size_t)NB * 2 * CC;
  float* hh     = ws + off; off += (size_t)NB * CC;
  (void)ws_size;

  float* out = (float*)d_out;   // [0:128) po, [128:256) go, [256:384) attn, [384:512) comb

  // protein branch
  run_type(x_p, ei_p, ei_p + E, batch_p, NPn, E,
           P[6], P[7], P[10], P[11],
           P[14], P[15], P[16], P[20], P[21], P[22],
           P[26], P[27], P[28], P[29], P[30], P[31],
           h1, xw, agg, dinv, sum, cnt, pooled, hh, out, stream);
  // gene branch
  run_type(x_g, ei_g, ei_g + E, batch_g, NGn, E,
           P[8], P[9], P[12], P[13],
           P[17], P[18], P[19], P[23], P[24], P[25],
           P[32], P[33], P[34], P[35], P[36], P[37],
           h1, xw, agg, dinv, sum, cnt, pooled, hh, out + 128, stream);

  // attention + combine
  k_attention<<<1, NB, 0, stream>>>(out, P[38], P[39]);
}